// Multi_class_TCNN_output_layer_18717467476247
// MI455X (gfx1250) — compile-verified
//
#include <hip/hip_runtime.h>
#include <hip/hip_bf16.h>

// Problem constants (fixed by the reference)
#define NB      8
#define NN      512
#define HIST    24
#define SKIP    256
#define ENDC    128
#define KERN    13
#define FUT     12          // HIST - KERN + 1
#define OUTD    2
#define NODES   (NB * NN)   // 4096

// LDS tile strides (padded for bank-conflict-free ds_load_b64 / stores)
#define XS_STRIDE 260       // 256 data + 4 pad (keeps 16B alignment, spreads banks)
#define HS_STRIDE 132       // 128 data + 4 pad

typedef float v2f __attribute__((ext_vector_type(2)));
typedef float v8f __attribute__((ext_vector_type(8)));

__global__ __launch_bounds__(256)
void moe_tcnn_kernel(const float* __restrict__ x,        // (B,N,24,256)
                     const int*   __restrict__ labels,   // (B,N)
                     const float* __restrict__ W1,       // (8,128,256)
                     const float* __restrict__ b1,       // (8,128)
                     const float* __restrict__ W2,       // (8,2,128,13)
                     const float* __restrict__ b2,       // (8,2)
                     float*       __restrict__ out)      // (B,N,12,2)
{
    __shared__ float Xs[32 * XS_STRIDE];   // relu(x) tile, rows 24..31 zeroed (M pad)
    __shared__ float Hs[HIST * HS_STRIDE]; // stage-1 activations

    const int node = blockIdx.x;           // 0..4095 == b*N + n
    const int tid  = threadIdx.x;
    const int g    = labels[node];         // expert id

    // ---- Stage 0: load relu(X) into LDS (float4, fully coalesced) ----
    const float* xp = x + (size_t)node * (HIST * SKIP);
    #pragma unroll
    for (int q = 0; q < 6; ++q) {          // 1536 float4 = 6144 floats / 256 threads
        int idx4 = tid + q * 256;
        int idx  = idx4 * 4;
        int row  = idx >> 8;               // /256
        int col  = idx & 255;
        float4 v = *(const float4*)(xp + idx);
        v.x = v.x > 0.f ? v.x : 0.f;
        v.y = v.y > 0.f ? v.y : 0.f;
        v.z = v.z > 0.f ? v.z : 0.f;
        v.w = v.w > 0.f ? v.w : 0.f;
        *(float4*)(&Xs[row * XS_STRIDE + col]) = v;
    }
    // zero M-padding rows 24..31 (data columns only)
    #pragma unroll
    for (int q = 0; q < 8; ++q) {
        int i   = tid + q * 256;           // 8*256 elements
        int row = 24 + (i >> 8);
        int col = i & 255;
        Xs[row * XS_STRIDE + col] = 0.f;
    }
    __syncthreads();

    // ---- Stage 1: H(32x128) = relu(Xs(32x256) @ W1[g]^T + b1[g]) via WMMA f32 16x16x4 ----
    const int wv   = tid >> 5;             // wave id 0..7 -> N-strip
    const int lane = tid & 31;
    const int half = lane >> 4;            // 0: lanes 0-15, 1: lanes 16-31
    const int l16  = lane & 15;
    const int n0   = wv * 16;

    // B fragment source: W1[g][n0+l16][k + 2*half .. +1]  (float2, 8B aligned)
    const float* W1row = W1 + ((size_t)g * ENDC + n0 + l16) * SKIP + 2 * half;
    // A fragment sources (LDS): rows l16 (M-tile 0) and 16+l16 (M-tile 1)
    const float* Arow0 = &Xs[l16 * XS_STRIDE + 2 * half];
    const float* Arow1 = &Xs[(16 + l16) * XS_STRIDE + 2 * half];

    v8f c0 = {0.f,0.f,0.f,0.f,0.f,0.f,0.f,0.f};
    v8f c1 = {0.f,0.f,0.f,0.f,0.f,0.f,0.f,0.f};

    for (int kk = 0; kk < SKIP; kk += 32) {
        #pragma unroll
        for (int u = 0; u < 8; ++u) {
            int k = kk + u * 4;
            v2f af0 = *(const v2f*)(Arow0 + k);
            v2f af1 = *(const v2f*)(Arow1 + k);
            v2f bf  = *(const v2f*)(W1row + k);
            c0 = __builtin_amdgcn_wmma_f32_16x16x4_f32(false, af0, false, bf,
                                                       (short)0, c0, false, false);
            c1 = __builtin_amdgcn_wmma_f32_16x16x4_f32(false, af1, false, bf,
                                                       (short)0, c1, false, false);
        }
    }

    // bias + relu, scatter C tiles to Hs. C layout: VGPR j -> row (j + 8*half), col l16.
    const float bias = b1[g * ENDC + n0 + l16];
    const int   colH = n0 + l16;
    #pragma unroll
    for (int j = 0; j < 8; ++j) {
        int row0 = j + 8 * half;               // 0..15 always valid
        float v0 = c0[j] + bias;
        Hs[row0 * HS_STRIDE + colH] = v0 > 0.f ? v0 : 0.f;
        int row1 = 16 + j + 8 * half;          // 16..31, keep < 24
        if (row1 < HIST) {
            float v1 = c1[j] + bias;
            Hs[row1 * HS_STRIDE + colH] = v1 > 0.f ? v1 : 0.f;
        }
    }
    __syncthreads();

    // ---- Stage 2: out[f][o] = b2 + sum_{k,e} Hs[f+k][e] * W2[g][o][e][k] ----
    // 8-lane groups: group = wv*4 + lane/8  (32 groups, 24 used), l8 = lane%8
    const int group = wv * 4 + (lane >> 3);
    const int l8    = lane & 7;
    if (group < FUT * OUTD) {
        const int f = group % FUT;
        const int o = group / FUT;
        const float* W2go = W2 + ((size_t)(g * OUTD + o) * ENDC) * KERN; // [e][k]
        float acc = 0.f;
        #pragma unroll
        for (int i = 0; i < ENDC / 8; ++i) {   // 16 e-values per lane
            int e = l8 + 8 * i;
            const float* hcol = &Hs[f * HS_STRIDE + e];
            const float* wrow = &W2go[e * KERN];
            #pragma unroll
            for (int k = 0; k < KERN; ++k)
                acc += hcol[k * HS_STRIDE] * wrow[k];
        }
        // reduce over the 8 lanes of the group (groups are 8-lane aligned in wave32)
        acc += __shfl_xor(acc, 4, 32);
        acc += __shfl_xor(acc, 2, 32);
        acc += __shfl_xor(acc, 1, 32);
        if (l8 == 0)
            out[(size_t)node * (FUT * OUTD) + f * OUTD + o] = acc + b2[g * OUTD + o];
    }
}

extern "C" void kernel_launch(void* const* d_in, const int* in_sizes, int n_in,
                              void* d_out, int out_size, void* d_ws, size_t ws_size,
                              hipStream_t stream) {
    const float* x      = (const float*)d_in[0];
    const int*   labels = (const int*)  d_in[1];
    const float* W1     = (const float*)d_in[2];
    const float* b1     = (const float*)d_in[3];
    const float* W2     = (const float*)d_in[4];
    const float* b2     = (const float*)d_in[5];
    float*       out    = (float*)d_out;

    moe_tcnn_kernel<<<NODES, 256, 0, stream>>>(x, labels, W1, b1, W2, b2, out);
}